// OurModel_34110630265638
// MI455X (gfx1250) — compile-verified
//
#include <hip/hip_runtime.h>
#include <hip/hip_bf16.h>
#include <math.h>

// ---------------------------------------------------------------------------
// Model constants (from reference)
// ---------------------------------------------------------------------------
#define DHID    128
#define NNODES  50000
#define NEDGES  600000
#define BATCH   16
#define SEQ     512
#define NHEADS  4
#define HDIM    32      // DHID / NHEADS

typedef __attribute__((ext_vector_type(2))) float v2f;
typedef __attribute__((ext_vector_type(8))) float v8f;

// ---------------------------------------------------------------------------
// float atomic max via ordered-int trick (m initialized to -inf)
// ---------------------------------------------------------------------------
__device__ __forceinline__ void atomicMaxFloat(float* addr, float val) {
  if (val >= 0.0f)
    atomicMax((int*)addr, __float_as_int(val));
  else
    atomicMin((unsigned int*)addr, __float_as_uint(val));
}

// ---------------------------------------------------------------------------
// WMMA GEMM: out[M x 128] = A[M x 128] @ W[128 x 128]  (+bias) (+epilogue)
//   mode 0: out = acc + bias            (bias may be nullptr)
//   mode 1: out = R[row] + relu(acc + bias)
// One wave computes one 16x16 output tile with V_WMMA_F32_16X16X4_F32,
// K-loop of 32 steps. M must be a multiple of 16 (50000 and 8192 both are).
//
// Fragment layouts per CDNA5 ISA 7.12.2 (f32):
//   A 16x4 : lane l<16 -> row=l, VGPR0=K0,VGPR1=K1 ; lane l+16 -> K2,K3
//   B 4x16 : lane l<16 -> col=l, VGPR0=K0,VGPR1=K1 ; lane l+16 -> K2,K3
//   C/D    : VGPR r -> M=r (lanes 0..15), M=r+8 (lanes 16..31), N=lane&15
// ---------------------------------------------------------------------------
__global__ __launch_bounds__(128)
void gemm_n128_kernel(const float* __restrict__ A, const float* __restrict__ W,
                      const float* __restrict__ bias, const float* __restrict__ R,
                      float* __restrict__ out, int M, int mode) {
  int wave = blockIdx.x * (blockDim.x >> 5) + (threadIdx.x >> 5);
  int lane = threadIdx.x & 31;
  int tile_m = wave >> 3;       // 8 column tiles (N=128)
  int tile_n = wave & 7;
  if (tile_m * 16 >= M) return; // wave-uniform: EXEC stays all-ones for WMMA
  int row0 = tile_m * 16, col0 = tile_n * 16;
  int l    = lane & 15;
  int half = lane >> 4;
  int ksel = half << 1;         // 0 or 2

  float accv[8];
#if defined(__AMDGCN__) && __has_builtin(__builtin_amdgcn_wmma_f32_16x16x4_f32)
  v8f acc = {};
  const float* arow = A + (size_t)(row0 + l) * DHID;
  const float* wcol = W + col0 + l;
#pragma unroll
  for (int k = 0; k < DHID; k += 4) {
    v2f a, b;
    a.x = arow[k + ksel];
    a.y = arow[k + ksel + 1];
    b.x = wcol[(size_t)(k + ksel) * DHID];
    b.y = wcol[(size_t)(k + ksel + 1) * DHID];
    acc = __builtin_amdgcn_wmma_f32_16x16x4_f32(false, a, false, b,
                                                (short)0, acc, false, false);
  }
#pragma unroll
  for (int r = 0; r < 8; ++r) accv[r] = acc[r];
#else
  // scalar fallback (host pass / missing builtin)
#pragma unroll
  for (int r = 0; r < 8; ++r) accv[r] = 0.0f;
  for (int r = 0; r < 8; ++r) {
    const float* arow = A + (size_t)(row0 + r + 8 * half) * DHID;
    const float* wcol = W + col0 + l;
    float s = 0.0f;
    for (int k = 0; k < DHID; ++k) s += arow[k] * wcol[(size_t)k * DHID];
    accv[r] = s;
  }
#endif
#pragma unroll
  for (int r = 0; r < 8; ++r) {
    int rr = row0 + r + 8 * half;
    int cc = col0 + l;
    float v = accv[r];
    if (bias) v += bias[cc];
    if (mode == 1) {
      v = v > 0.0f ? v : 0.0f;
      v += R[(size_t)rr * DHID + cc];
    }
    out[(size_t)rr * DHID + cc] = v;
  }
}

// ---------------------------------------------------------------------------
// GAT: a_s[n] = xw[n].att_src ; a_d[n] = xw[n].att_dst
// ---------------------------------------------------------------------------
__global__ void att_scores_kernel(const float* __restrict__ xw,
                                  const float* __restrict__ asrc,
                                  const float* __restrict__ adst,
                                  float* __restrict__ a_s, float* __restrict__ a_d,
                                  int N) {
  int n = blockIdx.x * blockDim.x + threadIdx.x;
  if (n >= N) return;
  const float* row = xw + (size_t)n * DHID;
  float s = 0.f, d = 0.f;
  for (int e = 0; e < DHID; ++e) { float x = row[e]; s += x * asrc[e]; d += x * adst[e]; }
  a_s[n] = s; a_d[n] = d;
}

__global__ void gat_init_kernel(float* xacc, float* m, float* den, int NH, int N) {
  int i = blockIdx.x * blockDim.x + threadIdx.x;
  if (i < NH) xacc[i] = 0.0f;
  if (i < N) { m[i] = -INFINITY; den[i] = 0.0f; }
}

// pass 1: e = leaky_relu(a_s[src]+a_d[dst], 0.2); segment-max into m[dst]
__global__ void edge_pass1_kernel(const int* __restrict__ src, const int* __restrict__ dst,
                                  const float* __restrict__ a_s, const float* __restrict__ a_d,
                                  float* __restrict__ ebuf, float* __restrict__ m,
                                  int E, int N) {
  int i = blockIdx.x * blockDim.x + threadIdx.x;
  if (i >= E + N) return;
  int s, d;
  if (i < E) { s = src[i]; d = dst[i]; } else { s = d = i - E; } // self loops
  float e = a_s[s] + a_d[d];
  e = e > 0.0f ? e : 0.2f * e;
  ebuf[i] = e;
  atomicMaxFloat(&m[d], e);
}

// pass 2: ex = exp(e - m[dst]); segment-sum into den[dst]
__global__ void edge_pass2_kernel(const int* __restrict__ dst, float* __restrict__ ebuf,
                                  const float* __restrict__ m, float* __restrict__ den,
                                  int E, int N) {
  int i = blockIdx.x * blockDim.x + threadIdx.x;
  if (i >= E + N) return;
  int d = (i < E) ? dst[i] : (i - E);
  float ex = __expf(ebuf[i] - m[d]);
  ebuf[i] = ex;
  atomicAdd(&den[d], ex);
}

// pass 3: xacc[dst] += (ex/den[dst]) * xw[src]  -- one wave per edge, float4/lane
__global__ __launch_bounds__(256)
void edge_pass3_kernel(const int* __restrict__ src, const int* __restrict__ dst,
                       const float* __restrict__ ebuf, const float* __restrict__ den,
                       const float* __restrict__ xw, float* __restrict__ xacc,
                       int E, int N) {
  int wid  = blockIdx.x * (blockDim.x >> 5) + (threadIdx.x >> 5);
  int lane = threadIdx.x & 31;
  if (wid >= E + N) return;
  int s, d;
  if (wid < E) { s = src[wid]; d = dst[wid]; } else { s = d = wid - E; }
  float coef = ebuf[wid] / den[d];
  const float4* xr = (const float4*)(xw + (size_t)s * DHID);
  float4 v = xr[lane];
  float* o = xacc + (size_t)d * DHID + lane * 4;
  atomicAdd(o + 0, coef * v.x);
  atomicAdd(o + 1, coef * v.y);
  atomicAdd(o + 2, coef * v.z);
  atomicAdd(o + 3, coef * v.w);
}

// X[r,:] = xacc[node_ids[r],:] + gat_bias
__global__ __launch_bounds__(128)
void gather_kernel(const int* __restrict__ node_ids, const float* __restrict__ xacc,
                   const float* __restrict__ gbias, float* __restrict__ X) {
  int r = blockIdx.x, t = threadIdx.x;
  int n = node_ids[r];
  X[(size_t)r * DHID + t] = xacc[(size_t)n * DHID + t] + gbias[t];
}

// ---------------------------------------------------------------------------
// MAB with 1-token Q (mab0/2/4):  H[b] = O + relu(O@Wo+bo),
//   Qp = seed@Wq+bq (1x128), scores over 512 keys x 4 heads, O = Qp + A@Vp.
// One workgroup (128 threads) per batch sample.
// ---------------------------------------------------------------------------
__global__ __launch_bounds__(128)
void mab_front_attn_kernel(const float* __restrict__ seed,
                           const float* __restrict__ Wq, const float* __restrict__ bq,
                           const float* __restrict__ Kp, const float* __restrict__ Vp,
                           const float* __restrict__ Wo, const float* __restrict__ bo,
                           float* __restrict__ outH) {
  __shared__ float Qp[DHID];
  __shared__ float sc[NHEADS * SEQ];
  __shared__ float O[DHID];
  int b = blockIdx.x, t = threadIdx.x;
  // Qp (same for all samples; recomputed per workgroup — 128 FMAs/thread)
  float q = bq[t];
  for (int e = 0; e < DHID; ++e) q += seed[e] * Wq[(size_t)e * DHID + t];
  Qp[t] = q;
  __syncthreads();
  const float scale = 0.0883883476483184f; // 1/sqrt(128), full-dim scale per reference
  for (int k = t; k < SEQ; k += 128) {
    const float* krow = Kp + (size_t)(b * SEQ + k) * DHID;
    for (int h = 0; h < NHEADS; ++h) {
      float s = 0.f;
      for (int d = 0; d < HDIM; ++d) s += Qp[h * HDIM + d] * krow[h * HDIM + d];
      sc[h * SEQ + k] = s * scale;
    }
  }
  __syncthreads();
  if (t < NHEADS) {
    int h = t;
    float mx = -INFINITY;
    for (int k = 0; k < SEQ; ++k) mx = fmaxf(mx, sc[h * SEQ + k]);
    float sum = 0.f;
    for (int k = 0; k < SEQ; ++k) { float ex = __expf(sc[h * SEQ + k] - mx); sc[h * SEQ + k] = ex; sum += ex; }
    float inv = 1.0f / sum;
    for (int k = 0; k < SEQ; ++k) sc[h * SEQ + k] *= inv;
  }
  __syncthreads();
  {
    int h = t >> 5; // head covering column t
    float o = Qp[t];
    const float* vcol = Vp + (size_t)b * SEQ * DHID + t;
    const float* a = sc + h * SEQ;
    for (int k = 0; k < SEQ; ++k) o += a[k] * vcol[(size_t)k * DHID];
    O[t] = o;
  }
  __syncthreads();
  float y = bo[t];
  for (int e = 0; e < DHID; ++e) y += O[e] * Wo[(size_t)e * DHID + t];
  y = y > 0.0f ? y : 0.0f;
  outH[(size_t)b * DHID + t] = O[t] + y;
}

// Y[b] = Xin[b] @ W + bias   (16 samples, 1x128 each)
__global__ __launch_bounds__(128)
void matvec16_kernel(const float* __restrict__ Xin, const float* __restrict__ W,
                     const float* __restrict__ bias, float* __restrict__ Y) {
  __shared__ float Xl[DHID];
  int b = blockIdx.x, t = threadIdx.x;
  Xl[t] = Xin[(size_t)b * DHID + t];
  __syncthreads();
  float y = bias[t];
  for (int e = 0; e < DHID; ++e) y += Xl[e] * W[(size_t)e * DHID + t];
  Y[(size_t)b * DHID + t] = y;
}

// O[row,:] += Vp[row/512,:]   (broadcast 1-token V across the 512 queries)
__global__ void add_bcast_kernel(float* __restrict__ O, const float* __restrict__ Vp, int total) {
  int idx = blockIdx.x * blockDim.x + threadIdx.x;
  if (idx >= total) return;
  int t = idx & (DHID - 1);
  int b = idx >> 16;        // idx / (128*512)
  O[idx] += Vp[b * DHID + t];
}

// SAB on length-1 set: softmax(1 key)=1 -> O = Qp+Vp; out = O + relu(O@Wo+bo)
__global__ __launch_bounds__(128)
void sab1_kernel(const float* __restrict__ Xin,
                 const float* __restrict__ Wq, const float* __restrict__ bq,
                 const float* __restrict__ Wv, const float* __restrict__ bv,
                 const float* __restrict__ Wo, const float* __restrict__ bo,
                 float* __restrict__ Xout) {
  __shared__ float Xl[DHID];
  __shared__ float O[DHID];
  int b = blockIdx.x, t = threadIdx.x;
  Xl[t] = Xin[(size_t)b * DHID + t];
  __syncthreads();
  float qp = bq[t], vp = bv[t];
  for (int e = 0; e < DHID; ++e) {
    float xe = Xl[e];
    qp += xe * Wq[(size_t)e * DHID + t];
    vp += xe * Wv[(size_t)e * DHID + t];
  }
  O[t] = qp + vp;
  __syncthreads();
  float y = bo[t];
  for (int e = 0; e < DHID; ++e) y += O[e] * Wo[(size_t)e * DHID + t];
  y = y > 0.0f ? y : 0.0f;
  Xout[(size_t)b * DHID + t] = O[t] + y;
}

// y = X@dec_W+dec_b -> out[b,0,:] ; out[b,1+p,:] = prototypes[p,:]
__global__ __launch_bounds__(128)
void dec_out_kernel(const float* __restrict__ Xin, const float* __restrict__ W,
                    const float* __restrict__ bias, const float* __restrict__ proto,
                    float* __restrict__ out) {
  __shared__ float Xl[DHID];
  int b = blockIdx.x, t = threadIdx.x;
  Xl[t] = Xin[(size_t)b * DHID + t];
  __syncthreads();
  float y = bias[t];
  for (int e = 0; e < DHID; ++e) y += Xl[e] * W[(size_t)e * DHID + t];
  float* ob = out + (size_t)b * 6 * DHID;
  ob[t] = y;
  for (int p = 0; p < 5; ++p) ob[(size_t)(1 + p) * DHID + t] = proto[(size_t)p * DHID + t];
}

// ---------------------------------------------------------------------------
// Orchestration. Workspace need: ~14.71M floats (~59 MB).
// ---------------------------------------------------------------------------
extern "C" void kernel_launch(void* const* d_in, const int* in_sizes, int n_in,
                              void* d_out, int out_size, void* d_ws, size_t ws_size,
                              hipStream_t stream) {
  (void)in_sizes; (void)n_in; (void)out_size; (void)ws_size;
  const int*   node_ids   = (const int*)  d_in[0];
  const int*   edge_idx   = (const int*)  d_in[1];
  /* d_in[2] = edge_attr (unused by reference GATConv) */
  const float* node_embed = (const float*)d_in[3];
  const float* lin_W  = (const float*)d_in[4];
  const float* lin_b  = (const float*)d_in[5];
  const float* gat_W  = (const float*)d_in[6];
  const float* gat_as = (const float*)d_in[7];
  const float* gat_ad = (const float*)d_in[8];
  const float* gat_b  = (const float*)d_in[9];
  const float* mab_Wq = (const float*)d_in[10];
  const float* mab_bq = (const float*)d_in[11];
  const float* mab_Wk = (const float*)d_in[12];
  const float* mab_bk = (const float*)d_in[13];
  const float* mab_Wv = (const float*)d_in[14];
  const float* mab_bv = (const float*)d_in[15];
  const float* mab_Wo = (const float*)d_in[16];
  const float* mab_bo = (const float*)d_in[17];
  const float* isab_I = (const float*)d_in[18];
  const float* pma_S  = (const float*)d_in[19];
  const float* dec_W  = (const float*)d_in[20];
  const float* dec_b  = (const float*)d_in[21];
  const float* proto  = (const float*)d_in[22];
  float* out = (float*)d_out;

  // workspace layout (floats)
  float* f    = (float*)d_ws;
  float* buf1 = f;                                   // 6.40M : x0 -> xacc -> Kp/Qp/O
  float* buf2 = buf1 + (size_t)NNODES * DHID;        // 6.40M : xw -> Vp
  float* X    = buf2 + (size_t)NNODES * DHID;        // 1.05M : gathered set [16,512,128]
  float* a_s  = X + (size_t)BATCH * SEQ * DHID;      // 50000
  float* a_d  = a_s + NNODES;                        // 50000
  float* mbuf = a_d + NNODES;                        // 50000
  float* den  = mbuf + NNODES;                       // 50000
  float* ebuf = den + NNODES;                        // 650000
  float* Hb   = ebuf + (NEDGES + NNODES);            // 2048
  float* VpB  = Hb + BATCH * DHID;                   // 2048
  float* Xs   = VpB + BATCH * DHID;                  // 2048
  float* Xt   = Xs + BATCH * DHID;                   // 2048

  auto gemm = [&](const float* A, const float* W, const float* bias, const float* R,
                  float* outp, int M, int mode) {
    int waves  = (M / 16) * 8;          // M multiple of 16; N=128 -> 8 col tiles
    int blocks = (waves + 3) / 4;       // 4 waves / 128-thread block
    gemm_n128_kernel<<<blocks, 128, 0, stream>>>(A, W, bias, R, outp, M, mode);
  };

  // ---- GNN front end -----------------------------------------------------
  gemm(node_embed, lin_W, lin_b, nullptr, buf1, NNODES, 0);       // x0
  gemm(buf1, gat_W, nullptr, nullptr, buf2, NNODES, 0);           // xw
  att_scores_kernel<<<(NNODES + 255) / 256, 256, 0, stream>>>(buf2, gat_as, gat_ad,
                                                              a_s, a_d, NNODES);
  gat_init_kernel<<<(NNODES * DHID + 255) / 256, 256, 0, stream>>>(buf1, mbuf, den,
                                                                   NNODES * DHID, NNODES);
  const int* esrc = edge_idx;
  const int* edst = edge_idx + NEDGES;
  int ETOT = NEDGES + NNODES;
  edge_pass1_kernel<<<(ETOT + 255) / 256, 256, 0, stream>>>(esrc, edst, a_s, a_d,
                                                            ebuf, mbuf, NEDGES, NNODES);
  edge_pass2_kernel<<<(ETOT + 255) / 256, 256, 0, stream>>>(edst, ebuf, mbuf, den,
                                                            NEDGES, NNODES);
  edge_pass3_kernel<<<(ETOT + 7) / 8, 256, 0, stream>>>(esrc, edst, ebuf, den,
                                                        buf2, buf1, NEDGES, NNODES);
  gather_kernel<<<BATCH * SEQ, 128, 0, stream>>>(node_ids, buf1, gat_b, X);

  // ---- Encoder: 2 ISAB blocks (num_inds = 1) -----------------------------
  for (int i = 0; i < 2; ++i) {
    int m0 = 2 * i, m1 = 2 * i + 1;
    // front MAB: H = MAB(I_i, X)
    gemm(X, mab_Wk + (size_t)m0 * DHID * DHID, mab_bk + m0 * DHID, nullptr, buf1, BATCH * SEQ, 0);
    gemm(X, mab_Wv + (size_t)m0 * DHID * DHID, mab_bv + m0 * DHID, nullptr, buf2, BATCH * SEQ, 0);
    mab_front_attn_kernel<<<BATCH, 128, 0, stream>>>(isab_I + i * DHID,
        mab_Wq + (size_t)m0 * DHID * DHID, mab_bq + m0 * DHID, buf1, buf2,
        mab_Wo + (size_t)m0 * DHID * DHID, mab_bo + m0 * DHID, Hb);
    // back MAB: X = MAB(X, H)  (1 key -> softmax = 1, O = Qp + Vp)
    gemm(X, mab_Wq + (size_t)m1 * DHID * DHID, mab_bq + m1 * DHID, nullptr, buf1, BATCH * SEQ, 0);
    matvec16_kernel<<<BATCH, 128, 0, stream>>>(Hb, mab_Wv + (size_t)m1 * DHID * DHID,
                                               mab_bv + m1 * DHID, VpB);
    add_bcast_kernel<<<(BATCH * SEQ * DHID) / 256, 256, 0, stream>>>(buf1, VpB,
                                                                     BATCH * SEQ * DHID);
    gemm(buf1, mab_Wo + (size_t)m1 * DHID * DHID, mab_bo + m1 * DHID, buf1, X, BATCH * SEQ, 1);
  }

  // ---- Decoder: PMA (mab4) -> SAB (mab5) -> SAB (mab6) -> Linear ---------
  gemm(X, mab_Wk + (size_t)4 * DHID * DHID, mab_bk + 4 * DHID, nullptr, buf1, BATCH * SEQ, 0);
  gemm(X, mab_Wv + (size_t)4 * DHID * DHID, mab_bv + 4 * DHID, nullptr, buf2, BATCH * SEQ, 0);
  mab_front_attn_kernel<<<BATCH, 128, 0, stream>>>(pma_S,
      mab_Wq + (size_t)4 * DHID * DHID, mab_bq + 4 * DHID, buf1, buf2,
      mab_Wo + (size_t)4 * DHID * DHID, mab_bo + 4 * DHID, Xs);
  sab1_kernel<<<BATCH, 128, 0, stream>>>(Xs,
      mab_Wq + (size_t)5 * DHID * DHID, mab_bq + 5 * DHID,
      mab_Wv + (size_t)5 * DHID * DHID, mab_bv + 5 * DHID,
      mab_Wo + (size_t)5 * DHID * DHID, mab_bo + 5 * DHID, Xt);
  sab1_kernel<<<BATCH, 128, 0, stream>>>(Xt,
      mab_Wq + (size_t)6 * DHID * DHID, mab_bq + 6 * DHID,
      mab_Wv + (size_t)6 * DHID * DHID, mab_bv + 6 * DHID,
      mab_Wo + (size_t)6 * DHID * DHID, mab_bo + 6 * DHID, Xs);
  dec_out_kernel<<<BATCH, 128, 0, stream>>>(Xs, dec_W, dec_b, proto, out);
}